// SpookyNet_56684978372799
// MI455X (gfx1250) — compile-verified
//
#include <hip/hip_runtime.h>
#include <cstddef>

#define F_DIM 128
#define A_ATOMS 1024
#define B_MOL 32
#define N_ATOMS (B_MOL * A_ATOMS)
#define WPITCH 144  // float2 pitch: (2*144)%64==32 -> upper half-wave uses banks 32..63

typedef float v2f __attribute__((ext_vector_type(2)));
typedef float v8f __attribute__((ext_vector_type(8)));

__device__ __forceinline__ float fast_rcp(float x) {
  return __builtin_amdgcn_rcpf(x);  // v_rcp_f32
}
__device__ __forceinline__ float fast_rsq(float x) {
  return __builtin_amdgcn_rsqf(x);  // v_rsq_f32
}
__device__ __forceinline__ float silu_f(float x) {
  return x * fast_rcp(1.0f + __expf(-x));
}

// ---------------------------------------------------------------------------
// Kernel 0: tiny precompute.  u_plus = W_lin @ k_plus, u_minus = W_lin @ k_minus,
// wv = Wout @ w_e, c_plus = b_lin . k_plus, c_minus = b_lin . k_minus.
// ---------------------------------------------------------------------------
__global__ void precompute_kernel(const float* __restrict__ W_lin,
                                  const float* __restrict__ b_lin,
                                  const float* __restrict__ kp,
                                  const float* __restrict__ km,
                                  const float* __restrict__ Wout,
                                  const float* __restrict__ w_e,
                                  float* __restrict__ P) {
  const int t = threadIdx.x;  // 128 threads
  const float* wrow = W_lin + t * F_DIM;
  const float* orow = Wout + t * F_DIM;
  float up = 0.f, um = 0.f, wv = 0.f;
  for (int f = 0; f < F_DIM; ++f) {
    const float w = wrow[f];
    up += w * kp[f];
    um += w * km[f];
    wv += orow[f] * w_e[f];
  }
  P[t] = up;
  P[F_DIM + t] = um;
  P[2 * F_DIM + t] = wv;
  if (t == 0) {
    float cp = 0.f, cm = 0.f;
    for (int f = 0; f < F_DIM; ++f) {
      cp += b_lin[f] * kp[f];
      cm += b_lin[f] * km[f];
    }
    P[384] = cp;
    P[385] = cm;
  }
}

// ---------------------------------------------------------------------------
// Kernel 1: per-molecule attention weights a_i and pred_q init.
// ---------------------------------------------------------------------------
__global__ __launch_bounds__(1024) void attention_kernel(
    const float* __restrict__ e_z, const float* __restrict__ charge,
    const float* __restrict__ P, const float* __restrict__ w_e,
    const float* __restrict__ qz, const int* __restrict__ z,
    float* __restrict__ a_arr, float* __restrict__ pred) {
  __shared__ __align__(16) float su[F_DIM];
  __shared__ __align__(16) float swe[F_DIM];
  __shared__ float red[A_ATOMS];
  const int b = blockIdx.x;
  const int t = threadIdx.x;
  const float ch = charge[b];
  const bool posq = (ch >= 0.0f);
  if (t < F_DIM) {
    su[t] = posq ? P[t] : P[F_DIM + t];
    swe[t] = w_e[t];
  }
  __syncthreads();

  const int i = b * A_ATOMS + t;
  const float4* row = (const float4*)(e_z + (size_t)i * F_DIM);
  float dk = 0.f, dw = 0.f;
  const float4* su4 = (const float4*)su;
  const float4* swe4 = (const float4*)swe;
#pragma unroll 8
  for (int f4 = 0; f4 < F_DIM / 4; ++f4) {
    const float4 e = row[f4];
    const float4 u = su4[f4];
    const float4 w = swe4[f4];
    dk += e.x * u.x + e.y * u.y + e.z * u.z + e.w * u.w;
    dw += e.x * w.x + e.y * w.y + e.z * w.z + e.w * w.w;
  }
  const float c = posq ? P[384] : P[385];
  const float arg = (dk + c) * 0.08838834764831845f;  // 1/sqrt(128)
  const float num = (arg > 20.0f) ? arg : log1pf(__expf(arg));

  red[t] = num;
  __syncthreads();
  for (int s = A_ATOMS / 2; s > 0; s >>= 1) {
    if (t < s) red[t] += red[t + s];
    __syncthreads();
  }
  const float rden = fast_rcp(red[0]);

  a_arr[i] = ch * num * rden;
  pred[i] = dw + qz[z[i]];
}

// ---------------------------------------------------------------------------
// Kernel 2: h1 = silu(av) @ Wr1 via fp32 WMMA (16x16x4).
// B operand staged pair-interleaved in LDS: sWp[k/2][n] = (W[k,n], W[k+1,n]).
// Inner loop: batch-issue all 4 ds_load_2addr_b64, overlap with the silu
// exp/rcp chain, then 8 back-to-back WMMAs.
// ---------------------------------------------------------------------------
__global__ __launch_bounds__(256) void gemm1_kernel(
    const float* __restrict__ a_arr, const float* __restrict__ v_plus,
    const float* __restrict__ v_minus, const float* __restrict__ charge,
    const float* __restrict__ Wr1, float* __restrict__ h1) {
  __shared__ __align__(16) float2 sWp[32 * WPITCH];
  __shared__ __align__(16) float sv[F_DIM];
  const int tid = threadIdx.x;
  const int rowBase = blockIdx.x * 128;
  const bool pos = (charge[rowBase >> 10] >= 0.0f);
  const float* v = pos ? v_plus : v_minus;
  if (tid < F_DIM) sv[tid] = v[tid];

  const int lane = tid & 31;
  const int wave = tid >> 5;
  const int n0 = lane & 15;           // A: M, B/D: N
  const int khi = lane >> 4;          // this lane holds K pair (k0/2 + khi)
  const float am = a_arr[rowBase + wave * 16 + n0];

  v8f acc[8];
  const v8f vzero = {0.f, 0.f, 0.f, 0.f, 0.f, 0.f, 0.f, 0.f};
#pragma unroll
  for (int t = 0; t < 8; ++t) acc[t] = vzero;

  for (int half = 0; half < 2; ++half) {
    // stage 64 rows of Wr1, pair-interleaved
    const float4* src = (const float4*)(Wr1 + half * 64 * F_DIM);
    float* sWf = (float*)sWp;
    for (int idx = tid; idx < 64 * F_DIM / 4; idx += 256) {
      const float4 w4 = src[idx];
      const int k = idx >> 5;          // 0..63
      const int n = (idx & 31) << 2;   // 0..124
      const int base = ((k >> 1) * WPITCH + n) * 2 + (k & 1);
      sWf[base + 0] = w4.x;
      sWf[base + 2] = w4.y;
      sWf[base + 4] = w4.z;
      sWf[base + 6] = w4.w;
    }
    __syncthreads();

    const float2* brow = sWp + khi * WPITCH + n0;
#pragma unroll 2
    for (int k0 = 0; k0 < 64; k0 += 4) {
      // batch the independent LDS loads first
      float2 b2[8];
#pragma unroll
      for (int t = 0; t < 8; ++t) b2[t] = brow[t * 16];
      brow += 2 * WPITCH;
      // A fragment (exp/rcp latency overlaps the ds loads)
      const int kg = (half << 6) + k0 + (khi << 1);
      const float2 vp = *(const float2*)(sv + kg);
      v2f afr;
      afr.x = silu_f(am * vp.x);
      afr.y = silu_f(am * vp.y);
#pragma unroll
      for (int t = 0; t < 8; ++t) {
        v2f bfr;
        bfr.x = b2[t].x;
        bfr.y = b2[t].y;
        acc[t] = __builtin_amdgcn_wmma_f32_16x16x4_f32(
            false, afr, false, bfr, (short)0, acc[t], false, false);
      }
    }
    __syncthreads();
  }

  // D layout: VGPR r -> M = r + 8*khi; N = n0.
  const int rowHalf = khi << 3;
#pragma unroll
  for (int r = 0; r < 8; ++r) {
    float* out = h1 + (size_t)(rowBase + wave * 16 + rowHalf + r) * F_DIM + n0;
#pragma unroll
    for (int t = 0; t < 8; ++t) out[t * 16] = acc[t][r];
  }
}

// ---------------------------------------------------------------------------
// Kernel 3: h2 = silu(h1) @ Wr2 (fp32 WMMA), fused with the output projection:
// pred[i] += sum_j silu(a_i*v_j + h2[i,j]) * wv[j].
// ---------------------------------------------------------------------------
__global__ __launch_bounds__(256) void gemm2_kernel(
    const float* __restrict__ a_arr, const float* __restrict__ v_plus,
    const float* __restrict__ v_minus, const float* __restrict__ charge,
    const float* __restrict__ Wr2, const float* __restrict__ P,
    const float* __restrict__ h1, float* __restrict__ pred) {
  __shared__ __align__(16) float2 sWp[32 * WPITCH];
  __shared__ __align__(16) float sv[F_DIM];
  __shared__ __align__(16) float swv[F_DIM];
  __shared__ float sPred[128];
  const int tid = threadIdx.x;
  const int rowBase = blockIdx.x * 128;
  const bool pos = (charge[rowBase >> 10] >= 0.0f);
  const float* v = pos ? v_plus : v_minus;
  if (tid < F_DIM) {
    sv[tid] = v[tid];
    swv[tid] = P[2 * F_DIM + tid];
  }

  const int lane = tid & 31;
  const int wave = tid >> 5;
  const int n0 = lane & 15;
  const int khi = lane >> 4;
  const float* hrow = h1 + (size_t)(rowBase + wave * 16 + n0) * F_DIM;
  __builtin_prefetch(hrow, 0, 0);  // global_prefetch_b8

  v8f acc[8];
  const v8f vzero = {0.f, 0.f, 0.f, 0.f, 0.f, 0.f, 0.f, 0.f};
#pragma unroll
  for (int t = 0; t < 8; ++t) acc[t] = vzero;

  for (int half = 0; half < 2; ++half) {
    const float4* src = (const float4*)(Wr2 + half * 64 * F_DIM);
    float* sWf = (float*)sWp;
    for (int idx = tid; idx < 64 * F_DIM / 4; idx += 256) {
      const float4 w4 = src[idx];
      const int k = idx >> 5;
      const int n = (idx & 31) << 2;
      const int base = ((k >> 1) * WPITCH + n) * 2 + (k & 1);
      sWf[base + 0] = w4.x;
      sWf[base + 2] = w4.y;
      sWf[base + 4] = w4.z;
      sWf[base + 6] = w4.w;
    }
    __syncthreads();

    const float2* brow = sWp + khi * WPITCH + n0;
#pragma unroll 2
    for (int k0 = 0; k0 < 64; k0 += 4) {
      float2 b2[8];
#pragma unroll
      for (int t = 0; t < 8; ++t) b2[t] = brow[t * 16];
      brow += 2 * WPITCH;
      const int kg = (half << 6) + k0 + (khi << 1);
      const float2 hv = *(const float2*)(hrow + kg);  // global_load_b64
      v2f afr;
      afr.x = silu_f(hv.x);
      afr.y = silu_f(hv.y);
#pragma unroll
      for (int t = 0; t < 8; ++t) {
        v2f bfr;
        bfr.x = b2[t].x;
        bfr.y = b2[t].y;
        acc[t] = __builtin_amdgcn_wmma_f32_16x16x4_f32(
            false, afr, false, bfr, (short)0, acc[t], false, false);
      }
    }
    __syncthreads();
  }

  // Fused projection: lane holds rows (wave*16 + 8*khi + r), cols t*16+n0.
  const int rowHalf = khi << 3;
#pragma unroll
  for (int r = 0; r < 8; ++r) {
    const int mloc = wave * 16 + rowHalf + r;
    const float amr = a_arr[rowBase + mloc];
    float part = 0.f;
#pragma unroll
    for (int t = 0; t < 8; ++t) {
      const int n = t * 16 + n0;
      part += silu_f(amr * sv[n] + acc[t][r]) * swv[n];
    }
    // deterministic reduction across the 16 lanes that share this row
#pragma unroll
    for (int off = 8; off > 0; off >>= 1) part += __shfl_xor(part, off, 16);
    if (n0 == 0) sPred[mloc] = part;
  }
  __syncthreads();
  if (tid < 128) pred[rowBase + tid] += sPred[tid];
}

// ---------------------------------------------------------------------------
// Kernel 4: charge-conserving correction + O(A^2) switched Coulomb energy.
// Fast-transcendental inner loop: 2x v_rsq, 2x v_exp, 3x v_rcp per pair.
// ---------------------------------------------------------------------------
__global__ __launch_bounds__(1024) void electro_kernel(
    const float* __restrict__ xyz, const float* __restrict__ charge,
    const float* __restrict__ pred, float* __restrict__ out) {
  __shared__ __align__(16) float4 pq[A_ATOMS];
  __shared__ float red[A_ATOMS];
  const int b = blockIdx.x;
  const int t = threadIdx.x;
  const int i = b * A_ATOMS + t;

  const float p = pred[i];
  red[t] = p;
  __syncthreads();
  for (int s = A_ATOMS / 2; s > 0; s >>= 1) {
    if (t < s) red[t] += red[t + s];
    __syncthreads();
  }
  const float corr = (charge[b] - red[0]) * (1.0f / (float)A_ATOMS);
  const float qi = p + corr;

  const float* xp = xyz + (size_t)i * 3;
  float4 me;
  me.x = xp[0]; me.y = xp[1]; me.z = xp[2]; me.w = qi;
  pq[t] = me;
  __syncthreads();

  float e = 0.f;
#pragma unroll 4
  for (int j = 0; j < A_ATOMS; ++j) {
    const float4 o = pq[j];
    const float dx = me.x - o.x;
    const float dy = me.y - o.y;
    const float dz = me.z - o.z;
    if (j == t) continue;
    const float r2 = dx * dx + dy * dy + dz * dz;
    const float rinv = fast_rsq(r2);
    const float r = r2 * rinv;
    // f_switch: R_ON=1.25, R_OFF=3.75, 1/(R_OFF-R_ON)=0.4
    const float arg = (r - 1.25f) * 0.4f;
    const float omarg = 1.0f - arg;
    const float sa = (arg > 0.0f) ? __expf(-fast_rcp(arg)) : 0.0f;
    const float sb = (omarg > 0.0f) ? __expf(-fast_rcp(omarg)) : 0.0f;
    const float fs = sb * fast_rcp(sb + sa);
    const float coul = fs * fast_rsq(r2 + 1.0f) + (1.0f - fs) * rinv;
    e += o.w * coul;
  }
  e *= qi * 0.5f;  // each unordered pair counted twice

  __syncthreads();
  red[t] = e;
  __syncthreads();
  for (int s = A_ATOMS / 2; s > 0; s >>= 1) {
    if (t < s) red[t] += red[t + s];
    __syncthreads();
  }
  if (t == 0) out[b] = 332.0636f * red[0];
}

// ---------------------------------------------------------------------------
extern "C" void kernel_launch(void* const* d_in, const int* in_sizes, int n_in,
                              void* d_out, int out_size, void* d_ws,
                              size_t ws_size, hipStream_t stream) {
  (void)in_sizes; (void)n_in; (void)out_size; (void)ws_size;
  const float* e_z     = (const float*)d_in[0];
  const float* charge  = (const float*)d_in[1];
  const float* xyz     = (const float*)d_in[2];
  const float* W_lin   = (const float*)d_in[3];
  const float* b_lin   = (const float*)d_in[4];
  const float* k_plus  = (const float*)d_in[5];
  const float* k_minus = (const float*)d_in[6];
  const float* v_plus  = (const float*)d_in[7];
  const float* v_minus = (const float*)d_in[8];
  const float* Wr1     = (const float*)d_in[9];
  const float* Wr2     = (const float*)d_in[10];
  const float* Wout    = (const float*)d_in[11];
  const float* w_e     = (const float*)d_in[12];
  const float* qz      = (const float*)d_in[13];
  const int*   z       = (const int*)d_in[14];
  float* out = (float*)d_out;

  // workspace layout (floats): P[512] | a[N] | pred[N] | h1[N*128]
  float* P     = (float*)d_ws;
  float* a_arr = P + 512;
  float* pred  = a_arr + N_ATOMS;
  float* h1    = pred + N_ATOMS;

  precompute_kernel<<<1, 128, 0, stream>>>(W_lin, b_lin, k_plus, k_minus, Wout,
                                           w_e, P);
  attention_kernel<<<B_MOL, A_ATOMS, 0, stream>>>(e_z, charge, P, w_e, qz, z,
                                                  a_arr, pred);
  gemm1_kernel<<<N_ATOMS / 128, 256, 0, stream>>>(a_arr, v_plus, v_minus,
                                                  charge, Wr1, h1);
  gemm2_kernel<<<N_ATOMS / 128, 256, 0, stream>>>(a_arr, v_plus, v_minus,
                                                  charge, Wr2, P, h1, pred);
  electro_kernel<<<B_MOL, A_ATOMS, 0, stream>>>(xyz, charge, pred, out);
}